// GraphSimilarityModel_8349416423652
// MI455X (gfx1250) — compile-verified
//
#include <hip/hip_runtime.h>
#include <hip/hip_bf16.h>
#include <math.h>

// ---------------------------------------------------------------------------
// CDNA5 / gfx1250 implementation of the GraphSimilarityModel reference.
// Only g1 embeddings feed the outputs (reference bug preserved).
//
// GIN rewrite: (x + A x) @ W1 = Y + A·Y with Y = x@W1 (segment_sum is linear),
// so the big [N,d]@[d,64] GEMM reads X exactly once (271 MB total -> ~12 us at
// 23.3 TB/s) and all edge traffic is 64-wide. Big GEMMs use
// V_WMMA_F32_16X16X4_F32 (exact f32; op is HBM-bound, no reason to quantize).
// W1 K-panels are staged in LDS (row pitch 72 = bank-conflict-free for the
// half-wave K-split of the f32 WMMA B layout), cutting W L2 traffic 8x.
// ---------------------------------------------------------------------------

typedef __attribute__((ext_vector_type(2))) float v2f;
typedef __attribute__((ext_vector_type(8))) float v8f;

#define BN_SCALE 0.99999500003749971f   /* 1/sqrt(1+1e-5) */
#define SW_PITCH 72                     /* 64 cols + 8 pad floats */

// ---------------------------------------------------------------------------
// Y[N,64] = X[N,d] @ W[d,64]   (no bias)
// Block: 256 threads = 8 waves; each wave owns 32 rows (2 M-tiles x 4 N-tiles
// of 16x16 f32 WMMA accumulators -> 64 acc VGPRs, no spills).
// W is staged per 64-row K-chunk into LDS, zero-padded, so the hot loop has
// zero guards; only the final partial chunk guards the A (X) loads.
// Requires n_rows % 256 == 0 (N = 32768).
//
// WMMA f32 16x16x4 layouts (ISA 7.12.2):
//   A: lane<16 -> row=lane, {K,K+1}; lane>=16 -> row=lane-16, {K+2,K+3}
//   B: V0 low lanes = row K, V1 low = K+1, V0 high = K+2, V1 high = K+3
//   D: vgpr v, lane l: row = v + 8*(l>=16), col = l&15
// ---------------------------------------------------------------------------
__global__ __launch_bounds__(256) void gemm_x_w64(
    const float* __restrict__ X, const float* __restrict__ W,
    float* __restrict__ Y, int d) {
  __shared__ float sW[64 * SW_PITCH];

  const int lane = threadIdx.x & 31;
  const int wave = threadIdx.x >> 5;
  const int row0 = blockIdx.x * 256 + wave * 32;   // 32 rows per wave
  const int hi = lane >> 4;                        // 0: {K,K+1}  1: {K+2,K+3}
  const int ln = lane & 15;

  const float* xr0 = X + (size_t)(row0 + ln) * d;
  const float* xr1 = X + (size_t)(row0 + 16 + ln) * d;

  v8f acc[2][4];
#pragma unroll
  for (int i = 0; i < 2; ++i)
#pragma unroll
    for (int j = 0; j < 4; ++j) acc[i][j] = (v8f){0, 0, 0, 0, 0, 0, 0, 0};

  for (int kc = 0; kc < d; kc += 64) {
    const int rem = d - kc;   // > 0
    // Cooperative stage of W[kc:kc+64, 0:64] into LDS, zero-padded past d.
    for (int idx = threadIdx.x; idx < 64 * 64; idx += 256) {
      const int r = idx >> 6, c = idx & 63;
      sW[r * SW_PITCH + c] = (r < rem) ? W[(size_t)(kc + r) * 64 + c] : 0.f;
    }
    __syncthreads();

    if (rem >= 64) {
      // Hot path: fully unguarded.
#pragma unroll 4
      for (int k = 0; k < 64; k += 4) {
        const int ka = kc + k + hi * 2;
        v2f a0, a1;
        a0.x = xr0[ka]; a0.y = xr0[ka + 1];
        a1.x = xr1[ka]; a1.y = xr1[ka + 1];
        const float* s0 = &sW[(k + hi * 2) * SW_PITCH];
        const float* s1 = s0 + SW_PITCH;
#pragma unroll
        for (int nt = 0; nt < 4; ++nt) {
          v2f b;
          b.x = s0[nt * 16 + ln];
          b.y = s1[nt * 16 + ln];
          acc[0][nt] = __builtin_amdgcn_wmma_f32_16x16x4_f32(false, a0, false, b, (short)0, acc[0][nt], false, false);
          acc[1][nt] = __builtin_amdgcn_wmma_f32_16x16x4_f32(false, a1, false, b, (short)0, acc[1][nt], false, false);
        }
      }
    } else {
      // Tail chunk (d % 64 != 0): guard only the A loads; sW is zero-padded.
      for (int k = 0; k < rem; k += 4) {
        const int ka = kc + k + hi * 2;
        const bool g0 = (ka < d), g1 = (ka + 1 < d);
        v2f a0, a1;
        a0.x = g0 ? xr0[ka] : 0.f;  a0.y = g1 ? xr0[ka + 1] : 0.f;
        a1.x = g0 ? xr1[ka] : 0.f;  a1.y = g1 ? xr1[ka + 1] : 0.f;
        const float* s0 = &sW[(k + hi * 2) * SW_PITCH];
        const float* s1 = s0 + SW_PITCH;
#pragma unroll
        for (int nt = 0; nt < 4; ++nt) {
          v2f b;
          b.x = s0[nt * 16 + ln];
          b.y = s1[nt * 16 + ln];
          acc[0][nt] = __builtin_amdgcn_wmma_f32_16x16x4_f32(false, a0, false, b, (short)0, acc[0][nt], false, false);
          acc[1][nt] = __builtin_amdgcn_wmma_f32_16x16x4_f32(false, a1, false, b, (short)0, acc[1][nt], false, false);
        }
      }
    }
    __syncthreads();
  }

#pragma unroll
  for (int mt = 0; mt < 2; ++mt) {
    float* yout = Y + (size_t)(row0 + mt * 16) * 64;
#pragma unroll
    for (int v = 0; v < 8; ++v) {
      const int r = v + hi * 8;
#pragma unroll
      for (int nt = 0; nt < 4; ++nt)
        yout[(size_t)r * 64 + nt * 16 + ln] = acc[mt][nt][v];
    }
  }
}

// ---------------------------------------------------------------------------
// H[N,64] = relu( bn1( relu(Y + Z + b1) @ W2 + b2 ) )
// One wave per 16-row tile (d = 64 fixed, unguarded).
// ---------------------------------------------------------------------------
__global__ __launch_bounds__(256) void gemm_mid64(
    const float* __restrict__ Y, const float* __restrict__ Z,
    const float* __restrict__ b1, const float* __restrict__ W2,
    const float* __restrict__ b2, const float* __restrict__ bng,
    const float* __restrict__ bnb, float* __restrict__ H, int n_rows) {
  const int lane  = threadIdx.x & 31;
  const int wave  = threadIdx.x >> 5;
  const int mtile = blockIdx.x * (blockDim.x >> 5) + wave;
  const int row0  = mtile * 16;
  if (row0 >= n_rows) return;   // wave-uniform

  const int hi = lane >> 4;
  const int ln = lane & 15;
  const size_t abase = (size_t)(row0 + ln) * 64;

  v8f acc[4];
#pragma unroll
  for (int nt = 0; nt < 4; ++nt) acc[nt] = (v8f){0, 0, 0, 0, 0, 0, 0, 0};

#pragma unroll 4
  for (int k = 0; k < 64; k += 4) {
    const int ka = k + hi * 2;
    v2f a;
    a.x = fmaxf(Y[abase + ka]     + Z[abase + ka]     + b1[ka],     0.f);
    a.y = fmaxf(Y[abase + ka + 1] + Z[abase + ka + 1] + b1[ka + 1], 0.f);

    const float* wr0 = W2 + (size_t)ka * 64;
    const float* wr1 = wr0 + 64;
#pragma unroll
    for (int nt = 0; nt < 4; ++nt) {
      v2f b;
      b.x = wr0[nt * 16 + ln];
      b.y = wr1[nt * 16 + ln];
      acc[nt] = __builtin_amdgcn_wmma_f32_16x16x4_f32(false, a, false, b, (short)0, acc[nt], false, false);
    }
  }

#pragma unroll
  for (int nt = 0; nt < 4; ++nt) {
    const int col = nt * 16 + ln;
    const float sc = bng[col] * BN_SCALE;
    const float sh = bnb[col];
    const float bb = b2[col];
#pragma unroll
    for (int v = 0; v < 8; ++v) {
      const int r = v + hi * 8;
      float val = (acc[nt][v] + bb) * sc + sh;
      H[(size_t)(row0 + r) * 64 + col] = fmaxf(val, 0.f);
    }
  }
}

// ---------------------------------------------------------------------------
// accum[dst,:] += feat[src,:]  over all edges, 64-wide rows. thread=(edge,elem)
// ---------------------------------------------------------------------------
__global__ __launch_bounds__(256) void edge_scatter64(
    const float* __restrict__ feat, float* __restrict__ accum,
    const int* __restrict__ ei, int E) {
  const int t = blockIdx.x * blockDim.x + threadIdx.x;
  const int e = t >> 6;
  if (e >= E) return;
  const int j = t & 63;
  const int s = ei[e];       // ei[0] = src
  const int d = ei[E + e];   // ei[1] = dst
  atomicAdd(&accum[(size_t)d * 64 + j], feat[(size_t)s * 64 + j]);
}

// ---------------------------------------------------------------------------
// out8[n,8] = relu( bn2( relu((H+S2) @ w3 + b3) @ w4 + b4 ) ), one thread/node
// ---------------------------------------------------------------------------
__global__ __launch_bounds__(256) void head8(
    const float* __restrict__ H, const float* __restrict__ S2,
    const float* __restrict__ w3, const float* __restrict__ b3,
    const float* __restrict__ w4, const float* __restrict__ b4,
    const float* __restrict__ bng, const float* __restrict__ bnb,
    float* __restrict__ out8, int n_rows) {
  const int r = blockIdx.x * blockDim.x + threadIdx.x;
  if (r >= n_rows) return;
  float u[8];
#pragma unroll
  for (int j = 0; j < 8; ++j) u[j] = b3[j];
  const size_t base = (size_t)r * 64;
  for (int k = 0; k < 64; ++k) {
    const float hv = H[base + k] + S2[base + k];
#pragma unroll
    for (int j = 0; j < 8; ++j) u[j] = fmaf(hv, w3[k * 8 + j], u[j]);
  }
#pragma unroll
  for (int j = 0; j < 8; ++j) u[j] = fmaxf(u[j], 0.f);
#pragma unroll
  for (int j = 0; j < 8; ++j) {
    float v = b4[j];
#pragma unroll
    for (int i = 0; i < 8; ++i) v = fmaf(u[i], w4[i * 8 + j], v);
    v = v * (bng[j] * BN_SCALE) + bnb[j];
    out8[(size_t)r * 8 + j] = fmaxf(v, 0.f);
  }
}

// pooled[batch[n]*24 + fpi*8 + j] += out8[n*8 + j]
__global__ __launch_bounds__(256) void pool_add(
    const float* __restrict__ out8, const int* __restrict__ batch,
    float* __restrict__ pooled, int n_rows, int fpi) {
  const int t = blockIdx.x * blockDim.x + threadIdx.x;
  const int n = t >> 3;
  if (n >= n_rows) return;
  const int j = t & 7;
  atomicAdd(&pooled[(size_t)batch[n] * 24 + fpi * 8 + j], out8[(size_t)n * 8 + j]);
}

__global__ __launch_bounds__(256) void count_nodes(
    const int* __restrict__ batch, float* __restrict__ cnt, int n) {
  const int i = blockIdx.x * blockDim.x + threadIdx.x;
  if (i < n) atomicAdd(&cnt[batch[i]], 1.0f);
}

__global__ __launch_bounds__(256) void zero_f32(float* __restrict__ p, size_t n) {
  const size_t i = (size_t)blockIdx.x * blockDim.x + threadIdx.x;
  if (i < n) p[i] = 0.f;
}

// ---------------------------------------------------------------------------
// Attention + FC head, one thread per graph.
// out: [G] similarity, [G,8] comparison, [G,8] emb_u_agg (== emb_1_agg)
// ---------------------------------------------------------------------------
__global__ __launch_bounds__(256) void final_head(
    const float* __restrict__ pooled, const float* __restrict__ cnt,
    const float* __restrict__ att_w, const float* __restrict__ att_b,
    const float* __restrict__ fc1_w, const float* __restrict__ fc1_b,
    const float* __restrict__ fc1g, const float* __restrict__ fc1bb,
    const float* __restrict__ fc2_w, const float* __restrict__ fc2_b,
    const float* __restrict__ fc2g, const float* __restrict__ fc2bb,
    float* __restrict__ out, int G) {
  const int g = blockIdx.x * blockDim.x + threadIdx.x;
  if (g >= G) return;
  const float inv = 1.0f / fmaxf(cnt[g], 1.0f);

  float e[3][8], s[3];
#pragma unroll
  for (int i = 0; i < 3; ++i) {
    float sc = att_b[0];
#pragma unroll
    for (int j = 0; j < 8; ++j) {
      e[i][j] = pooled[(size_t)g * 24 + i * 8 + j] * inv;
      sc = fmaf(e[i][j], att_w[j], sc);
    }
    s[i] = sc;
  }
  float m = fmaxf(s[0], fmaxf(s[1], s[2]));
  float w0 = __expf(s[0] - m), w1 = __expf(s[1] - m), w2 = __expf(s[2] - m);
  const float wsum = w0 + w1 + w2;
  w0 /= wsum; w1 /= wsum; w2 /= wsum;

  float agg[8];
#pragma unroll
  for (int j = 0; j < 8; ++j)
    agg[j] = w0 * e[0][j] + w1 * e[1][j] + w2 * e[2][j];

  // combined = [agg, agg]
  float sl = fc1_b[0];
#pragma unroll
  for (int j = 0; j < 8; ++j) sl = fmaf(agg[j], fc1_w[j] + fc1_w[8 + j], sl);
  sl = sl * (fc1g[0] * BN_SCALE) + fc1bb[0];
  out[g] = 1.0f / (1.0f + __expf(-sl));

#pragma unroll
  for (int j = 0; j < 8; ++j) {
    float c = fc2_b[j];
#pragma unroll
    for (int i = 0; i < 8; ++i)
      c = fmaf(agg[i], fc2_w[i * 8 + j] + fc2_w[(8 + i) * 8 + j], c);
    c = c * (fc2g[j] * BN_SCALE) + fc2bb[j];
    out[(size_t)G + (size_t)g * 8 + j] = fmaxf(c, 0.f);
    out[(size_t)G * 9 + (size_t)g * 8 + j] = agg[j];
  }
}

// ---------------------------------------------------------------------------
// Host launcher.
// d_in layout (setup_inputs dict order, nested params flattened depth-first):
//  0 ei_g1[2,E] 1 batch_g1[N] 2..4 x_g1_{maccs,ecfp,pubchem}
//  5..9 g2, 10..14 gu (unused)
//  15+12*f: {w1,b1,w2,b2,w3,b3,w4,b4,bn1_g,bn1_b,bn2_g,bn2_b} for fp f
//  51 att_w 52 att_b 53 fc1_w 54 fc1_b 55 fc1_bn_g 56 fc1_bn_b
//  57 fc2_w 58 fc2_b 59 fc2_bn_g 60 fc2_bn_b  61 num_graphs
// ---------------------------------------------------------------------------
extern "C" void kernel_launch(void* const* d_in, const int* in_sizes, int n_in,
                              void* d_out, int out_size, void* d_ws, size_t ws_size,
                              hipStream_t stream) {
  (void)n_in; (void)ws_size;
  const int N = in_sizes[1];
  const int E = in_sizes[0] / 2;
  const int G = out_size / 17;                 // outputs: G + 8G + 8G
  const int dms[3] = { in_sizes[2] / N, in_sizes[3] / N, in_sizes[4] / N };

  const int* ei    = (const int*)d_in[0];
  const int* batch = (const int*)d_in[1];

  float* ws = (float*)d_ws;
  float* Y      = ws;                           // [N,64], also reused as S2
  float* Z      = ws + (size_t)N * 64;          // [N,64], also reused as out8
  float* H      = ws + (size_t)2 * N * 64;      // [N,64]
  float* pooled = ws + (size_t)3 * N * 64;      // [G,3,8]
  float* cnt    = pooled + (size_t)G * 24;      // [G]

  const size_t n64 = (size_t)N * 64;
  const size_t pz  = (size_t)G * 24 + (size_t)G;

  zero_f32<<<(unsigned)((pz + 255) / 256), 256, 0, stream>>>(pooled, pz);
  count_nodes<<<(N + 255) / 256, 256, 0, stream>>>(batch, cnt, N);

  const int blocksX = N / 256;                  // 256 rows per block (exact)
  const int wavesM  = (N + 15) / 16;            // 16 rows per wave (mid GEMM)
  const int blocksM = (wavesM + 7) / 8;
  const int eThreads = E * 64;

  for (int f = 0; f < 3; ++f) {
    const float* X   = (const float*)d_in[2 + f];
    const int pb = 15 + f * 12;
    const float* w1   = (const float*)d_in[pb + 0];
    const float* b1   = (const float*)d_in[pb + 1];
    const float* w2   = (const float*)d_in[pb + 2];
    const float* b2   = (const float*)d_in[pb + 3];
    const float* w3   = (const float*)d_in[pb + 4];
    const float* b3   = (const float*)d_in[pb + 5];
    const float* w4   = (const float*)d_in[pb + 6];
    const float* b4   = (const float*)d_in[pb + 7];
    const float* bn1g = (const float*)d_in[pb + 8];
    const float* bn1b = (const float*)d_in[pb + 9];
    const float* bn2g = (const float*)d_in[pb + 10];
    const float* bn2b = (const float*)d_in[pb + 11];

    // Y = X @ w1
    gemm_x_w64<<<blocksX, 256, 0, stream>>>(X, w1, Y, dms[f]);
    // Z = A . Y  (scatter-add over edges)
    zero_f32<<<(unsigned)((n64 + 255) / 256), 256, 0, stream>>>(Z, n64);
    edge_scatter64<<<(eThreads + 255) / 256, 256, 0, stream>>>(Y, Z, ei, E);
    // H = relu(bn1(relu(Y + Z + b1) @ w2 + b2))
    gemm_mid64<<<blocksM, 256, 0, stream>>>(Y, Z, b1, w2, b2, bn1g, bn1b, H, N);
    // S2 (reusing Y) = A . H
    zero_f32<<<(unsigned)((n64 + 255) / 256), 256, 0, stream>>>(Y, n64);
    edge_scatter64<<<(eThreads + 255) / 256, 256, 0, stream>>>(H, Y, ei, E);
    // out8 (reusing Z) = relu(bn2(relu((H + S2) @ w3 + b3) @ w4 + b4))
    head8<<<(N + 255) / 256, 256, 0, stream>>>(H, Y, w3, b3, w4, b4, bn2g, bn2b, Z, N);
    // pooled[:, f, :] += out8 grouped by batch
    pool_add<<<(N * 8 + 255) / 256, 256, 0, stream>>>(Z, batch, pooled, N, f);
  }

  final_head<<<(G + 255) / 256, 256, 0, stream>>>(
      pooled, cnt,
      (const float*)d_in[51], (const float*)d_in[52],
      (const float*)d_in[53], (const float*)d_in[54],
      (const float*)d_in[55], (const float*)d_in[56],
      (const float*)d_in[57], (const float*)d_in[58],
      (const float*)d_in[59], (const float*)d_in[60],
      (float*)d_out, G);
}